// DiffVolumeV2_34437047779565
// MI455X (gfx1250) — compile-verified
//
#include <hip/hip_runtime.h>
#include <stdint.h>

// Problem constants (from the reference setup).
#define B_   4
#define C_   32
#define H_   80
#define WL_  160
#define WR_  640
#define D_   48
#define THREADS_ 192   // 6 wave32s

typedef unsigned int u32x4 __attribute__((ext_vector_type(4)));
typedef unsigned int u32x8 __attribute__((ext_vector_type(8)));
typedef float        f32x4 __attribute__((ext_vector_type(4)));

// Issue one Tensor-Data-Mover load of a contiguous row of `n` fp32 elements
// from global memory into LDS at byte address `lds_byte_addr`.
// D# layout per CDNA5 ISA ch.8 (CDNA5 ISA):
//   group0: count=1 | lds_addr | global_addr[31:0] | global_addr[56:32], type=2
//   group1: data_size=4B | tensor_dim0=n | tensor_dim1=1 | tile_dim0=n |
//           tile_dim1=1 | tensor_dim0_stride=n
//   groups 2/3: zero (dims 2..4 unused)
__device__ __forceinline__ void tdm_load_row_f32(const float* gptr,
                                                 unsigned lds_byte_addr,
                                                 unsigned n) {
  unsigned long long ga = (unsigned long long)(uintptr_t)gptr;
  u32x4 g0;
  g0[0] = 1u;                                           // count = 1 (valid descriptor)
  g0[1] = lds_byte_addr;                                // lds_addr
  g0[2] = (unsigned)ga;                                 // global_addr[31:0]
  g0[3] = ((unsigned)(ga >> 32) & 0x01FFFFFFu)          // global_addr[56:32]
          | 0x80000000u;                                // type = 2 ("image")
  u32x8 g1;
  g1[0] = 2u << 16;                                     // data_size = 2 -> 4 bytes
  g1[1] = (n & 0xFFFFu) << 16;                          // tensor_dim0[15:0]
  g1[2] = 1u << 16;                                     // tensor_dim1 = 1
  g1[3] = (n & 0xFFFFu) << 16;                          // tile_dim0 = n
  g1[4] = 1u;                                           // tile_dim1 = 1 (tile_dim2 = 0)
  g1[5] = n;                                            // tensor_dim0_stride[31:0]
  g1[6] = 0u;                                           // stride hi / dim1_stride lo
  g1[7] = 0u;
  u32x4 gz = {0u, 0u, 0u, 0u};                          // groups 2 & 3: unused dims
  asm volatile("tensor_load_to_lds %0, %1, %2, %3"
               :: "s"(g0), "s"(g1), "s"(gz), "s"(gz)
               : "memory");
}

__global__ __launch_bounds__(THREADS_) void DiffVolumeV2_kernel(
    const float* __restrict__ left,   // [B, C, H, WL]
    const float* __restrict__ right,  // [B, C, H, WR]
    float* __restrict__ out)          // [B, C, D, H, WL]
{
  __shared__ float s_right[WR_];
  __shared__ float s_left[WL_];

  const int bch = blockIdx.x;       // (b*C + c)*H + h
  const int h   = bch % H_;
  const int bc  = bch / H_;

  // Wave 0 kicks off both async tensor loads and waits on TENSORcnt;
  // the workgroup barrier releases the other waves once LDS is populated.
  if (threadIdx.x < 32u) {
    tdm_load_row_f32(right + (size_t)bch * WR_,
                     (unsigned)(uintptr_t)&s_right[0], WR_);
    tdm_load_row_f32(left + (size_t)bch * WL_,
                     (unsigned)(uintptr_t)&s_left[0], WL_);
    __builtin_amdgcn_s_wait_tensorcnt(0);
  }
  __syncthreads();

  const size_t out_base = ((size_t)bc * D_ * H_ + h) * WL_;

  // 48 disparities x 40 float4-groups = 1920 vector stores; 10 per thread.
#pragma unroll
  for (int k = 0; k < (D_ * WL_ / 4) / THREADS_; ++k) {
    const int j  = (int)threadIdx.x + k * THREADS_;
    const int d  = j / (WL_ / 4);
    const int xg = (j - d * (WL_ / 4)) * 4;       // base x of this float4 group

    const f32x4 l = *(const f32x4*)&s_left[xg];

    // gather indices: clip(4x - d + 1, 0, WR-1); upper bound 637 never clips.
    const int r0 = 4 * xg - d + 1;
    const int i0 = (r0      < 0) ? 0 : r0;
    const int i1 = (r0 + 4  < 0) ? 0 : r0 + 4;
    const int i2 = (r0 + 8  < 0) ? 0 : r0 + 8;
    const int i3 = (r0 + 12 < 0) ? 0 : r0 + 12;

    f32x4 rv;
    rv[0] = s_right[i0];
    rv[1] = s_right[i1];
    rv[2] = s_right[i2];
    rv[3] = s_right[i3];

    const f32x4 o = l - rv;

    // Streaming store: output (315 MB) is written once, never re-read.
    __builtin_nontemporal_store(
        o, (f32x4*)(out + out_base + (size_t)d * (H_ * WL_) + xg));
  }
}

extern "C" void kernel_launch(void* const* d_in, const int* in_sizes, int n_in,
                              void* d_out, int out_size, void* d_ws, size_t ws_size,
                              hipStream_t stream) {
  (void)in_sizes; (void)n_in; (void)d_ws; (void)ws_size; (void)out_size;
  const float* left  = (const float*)d_in[0];
  const float* right = (const float*)d_in[1];
  // d_in[2] = max_disp scalar (48) — baked into the kernel constants.
  float* out = (float*)d_out;

  dim3 grid(B_ * C_ * H_);   // one workgroup per (b, c, h) row
  dim3 block(THREADS_);
  DiffVolumeV2_kernel<<<grid, block, 0, stream>>>(left, right, out);
}